// TDRouter_89962384982649
// MI455X (gfx1250) — compile-verified
//
#include <hip/hip_runtime.h>

typedef __attribute__((ext_vector_type(2))) float v2f;
typedef __attribute__((ext_vector_type(4))) float v4f;
typedef __attribute__((ext_vector_type(8))) float v8f;
typedef unsigned long long u64;
typedef unsigned int u32;

#define DIMQ   1024   // L == D == 1024
#define KSEL   256    // tokens kept
#define ROWS   16     // d-rows per workgroup
#define KSTEP  16     // l-tile per staging step
#define PITCH  20     // LDS pitch: conflict-free columns AND 16B-aligned rows
#define NITER  (DIMQ / KSTEP)

// Dynamic LDS layout (bytes):
//  GEMM phase : double-buffered {As[16][20] | Ws[1024][20]} = 2 x 83200
//  Sort phase : pairs[16][1024] u64 = 131072   (overlaps GEMM buffers)
//  Gather     : xcache[16][1024] f32 = 65536 at offset 131072
#define BUF_BYTES  (ROWS * PITCH * 4 + DIMQ * PITCH * 4)   // 83200
#define XCACHE_OFF (ROWS * DIMQ * 8)                       // 131072
#define SMEM_BYTES (XCACHE_OFF + ROWS * DIMQ * 4)          // 196608

#if defined(__AMDGCN__) && \
    __has_builtin(__builtin_amdgcn_global_load_async_to_lds_b128) && \
    __has_builtin(__builtin_amdgcn_global_load_async_to_lds_b32)
#define USE_ASYNC 1
typedef int v4i __attribute__((vector_size(16)));
typedef __attribute__((address_space(1))) int* gp32_t;
typedef __attribute__((address_space(3))) int* lp32_t;
typedef __attribute__((address_space(1))) v4i* gp128_t;
typedef __attribute__((address_space(3))) v4i* lp128_t;
#else
#define USE_ASYNC 0
#endif

__device__ __forceinline__ void wait_async0() {
#if USE_ASYNC
#if __has_builtin(__builtin_amdgcn_s_wait_asynccnt)
    __builtin_amdgcn_s_wait_asynccnt(0);
#else
    asm volatile("s_wait_asynccnt 0x0" ::: "memory");
#endif
#endif
}

// Stage one k-tile: As[d][l] = x[b, l0+l, d0+d];  Ws[m][l] = W[m, l0+l]
__device__ __forceinline__ void stage_tile(const float* __restrict__ xb,
                                           const float* __restrict__ W,
                                           char* buf, int l0, int d0, int tid) {
    float* As = (float*)buf;
    float* Ws = (float*)(buf + ROWS * PITCH * 4);
#if USE_ASYNC
    {
        int d = tid & 15, l = tid >> 4;
        __builtin_amdgcn_global_load_async_to_lds_b32(
            (gp32_t)(xb + (size_t)(l0 + l) * DIMQ + (d0 + d)),
            (lp32_t)(As + d * PITCH + l), 0, 0);
    }
    #pragma unroll
    for (int i = 0; i < 16; ++i) {
        int c = tid + 256 * i;        // 4096 x 16B chunks of the W tile
        int m = c >> 2, q = c & 3;
        __builtin_amdgcn_global_load_async_to_lds_b128(
            (gp128_t)(W + (size_t)m * DIMQ + l0 + q * 4),
            (lp128_t)(Ws + m * PITCH + q * 4), 0, 0);
    }
#else
    {
        int d = tid & 15, l = tid >> 4;
        As[d * PITCH + l] = xb[(size_t)(l0 + l) * DIMQ + (d0 + d)];
    }
    {
        int q = tid & 3, r = tid >> 2;
        #pragma unroll
        for (int p = 0; p < 16; ++p) {
            int m = r + p * 64;
            v4f wv = *(const v4f*)(W + (size_t)m * DIMQ + l0 + q * 4);
            *(v4f*)(Ws + m * PITCH + q * 4) = wv;   // 16B aligned (PITCH=20)
        }
    }
#endif
}

__global__ __launch_bounds__(256) void tdrouter_kernel(
    const float* __restrict__ x,      // (B, L, D)
    const float* __restrict__ W,      // (M, L) row-major
    const float* __restrict__ bias,   // (M)
    float* __restrict__ out)          // (B, K, D)
{
    extern __shared__ char smem[];
    u64*   pairs  = (u64*)smem;                    // 16 x 1024
    float* xcache = (float*)(smem + XCACHE_OFF);   // 16 x 1024

    const int tid  = threadIdx.x;
    const int lane = tid & 31;
    const int wave = tid >> 5;
    const int d0   = blockIdx.x * ROWS;
    const int b    = blockIdx.y;

    const float* xb = x + (size_t)b * DIMQ * DIMQ;

    v8f acc[8] = {};
    const int n0 = wave * 128;         // wave covers m in [n0, n0+128)
    const int M  = lane & 15;          // A row / B col within 16-tile
    const int kb = (lane >> 4) << 1;   // K sub-offset per lane half (f32 A/B layout)

    // ------------- GEMM: logits[d,m] = sum_l x[b,l,d] * W[m,l] -------------
    stage_tile(xb, W, smem, 0, d0, tid);           // prologue -> buf 0
    for (int it = 0; it < NITER; ++it) {
        wait_async0();
        __syncthreads();
        if (it + 1 < NITER)
            stage_tile(xb, W, smem + (((it + 1) & 1) ? BUF_BYTES : 0),
                       (it + 1) * KSTEP, d0, tid);

        const char*  buf = smem + ((it & 1) ? BUF_BYTES : 0);
        const float* As  = (const float*)buf;
        const float* Ws  = (const float*)(buf + ROWS * PITCH * 4);

        #pragma unroll
        for (int kk = 0; kk < KSTEP; kk += 4) {
            v2f afrag = *(const v2f*)(As + M * PITCH + kk + kb);   // 8B aligned
            #pragma unroll
            for (int j = 0; j < 8; ++j) {
                v2f bfrag = *(const v2f*)(Ws + (size_t)(n0 + j * 16 + M) * PITCH + kk + kb);
                acc[j] = __builtin_amdgcn_wmma_f32_16x16x4_f32(
                    false, afrag, false, bfrag, (short)0, acc[j], false, false);
            }
        }
    }
    __syncthreads();   // all GEMM LDS reads done before reusing the region

#if USE_ASYNC
    // Prefetch gather rows x[b, d0+r, :] into LDS; completes under the sort.
    #pragma unroll
    for (int i = 0; i < 16; ++i) {
        int c = tid + 256 * i;            // 4096 x 16B chunks
        int r = c >> 8;
        int pos = (c & 255) * 4;          // dword offset within row
        __builtin_amdgcn_global_load_async_to_lds_b128(
            (gp128_t)(xb + (size_t)(d0 + r) * DIMQ + pos),
            (lp128_t)(xcache + r * DIMQ + pos), 0, 0);
    }
#endif

    // -------- Epilogue: +bias, pack sortable (key, 1023-m) u64 into LDS --------
    {
        const int rbase = (lane >> 4) << 3;   // C/D: lanes 16-31 hold M = v + 8
        #pragma unroll
        for (int j = 0; j < 8; ++j) {
            int m = n0 + j * 16 + M;
            float bv = bias[m];
            #pragma unroll
            for (int v = 0; v < 8; ++v) {
                int drel = v + rbase;
                float val = acc[j][v] + bv;
                u32 u = __float_as_uint(val);
                u32 key = (u & 0x80000000u) ? ~u : (u | 0x80000000u);  // order-preserving
                u64 pk = ((u64)key << 32) | (u32)(DIMQ - 1 - m);       // tie -> lower m first
                pairs[(size_t)drel * DIMQ + m] = pk;
            }
        }
    }
    __syncthreads();

    // ---------------- Bitonic sort each row, descending ----------------
    for (u32 ks = 2; ks <= DIMQ; ks <<= 1) {
        for (u32 j = ks >> 1; j >= 1; j >>= 1) {
            #pragma unroll 1
            for (u32 s = 0; s < 32; ++s) {
                u32 id  = (u32)tid + 256u * s;          // 16 rows x 512 cexch
                u32 row = id >> 9;
                u32 q   = id & 511u;
                u32 lo  = q & (j - 1u);
                u32 i   = ((q ^ lo) << 1) | lo;
                u32 p   = i | j;
                u64* base = pairs + (size_t)row * DIMQ;
                u64 a = base[i];
                u64 c = base[p];
                bool sw = ((i & ks) == 0) ? (a < c) : (a > c);  // descending overall
                if (sw) { base[i] = c; base[p] = a; }
            }
            __syncthreads();
        }
    }

    wait_async0();       // xcache resident
    __syncthreads();

    // ---------------- Gather: out[b, k, d0+r] = x[b, d0+r, m_k] ----------------
    #pragma unroll 1
    for (int s = 0; s < 16; ++s) {
        int id = tid + 256 * s;           // 16 rows x 256 k
        int r  = id & 15;
        int k  = id >> 4;
        u64 pk = pairs[(size_t)r * DIMQ + k];
        int m  = (DIMQ - 1) - (int)(u32)(pk & 0xFFFFFFFFu);
#if USE_ASYNC
        float g = xcache[(size_t)r * DIMQ + m];
#else
        float g = xb[(size_t)(d0 + r) * DIMQ + m];
#endif
        out[((size_t)b * KSEL + k) * DIMQ + (d0 + r)] = g;
    }
}

extern "C" void kernel_launch(void* const* d_in, const int* in_sizes, int n_in,
                              void* d_out, int out_size, void* d_ws, size_t ws_size,
                              hipStream_t stream) {
    (void)in_sizes; (void)n_in; (void)d_ws; (void)ws_size; (void)out_size;
    const float* x    = (const float*)d_in[0];
    const float* W    = (const float*)d_in[1];
    const float* bias = (const float*)d_in[2];
    float* out        = (float*)d_out;

    dim3 grid(DIMQ / ROWS, 32);   // (64 d-tiles, B=32)
    dim3 block(256);
    tdrouter_kernel<<<grid, block, SMEM_BYTES, stream>>>(x, W, bias, out);
}